// SA_28063316312678
// MI455X (gfx1250) — compile-verified
//
#include <hip/hip_runtime.h>
#include <math.h>
#include <stdint.h>

#define D_ATT   256
#define D_GROUP 32
#define N_GROUP 8
#define BATCH   4
#define SEQ     512
#define M_TOT   (BATCH * SEQ)   // 2048 fused (b,l) rows

typedef __attribute__((ext_vector_type(2))) float v2f;
typedef __attribute__((ext_vector_type(8))) float v8f;

// ---------------------------------------------------------------------------
// Kernel 1: Q/K/V projections  y = x @ W^T  via V_WMMA_F32_16X16X4_F32.
// One wave per (matrix, 16x16 tile). K-dim = 256 -> 64 wmma steps.
// ---------------------------------------------------------------------------
__global__ __launch_bounds__(128) void qkv_proj_wmma(
    const float* __restrict__ x,
    const float* __restrict__ Wq,
    const float* __restrict__ Wk,
    const float* __restrict__ Wv,
    float* __restrict__ Q,
    float* __restrict__ K,
    float* __restrict__ V)
{
    const int lane = threadIdx.x & 31;
    const int wave = threadIdx.x >> 5;
    const int gwave = blockIdx.x * 4 + wave;     // 6144 waves total

    const int mat = gwave / (128 * 16);          // 0:Q 1:K 2:V
    const int rem = gwave % (128 * 16);
    const int mt  = rem >> 4;                    // 128 M-tiles
    const int nt  = rem & 15;                    // 16  N-tiles

    const float* __restrict__ W = (mat == 0) ? Wq : (mat == 1) ? Wk : Wv;
    float* __restrict__ Y       = (mat == 0) ? Q  : (mat == 1) ? K  : V;

    const int lm = lane & 15;                    // M-row (A) / N-col (B)
    const int h  = lane >> 4;                    // half select

    const float* __restrict__ arow = x + (size_t)(mt * 16 + lm) * D_ATT + 2 * h;
    const float* __restrict__ brow = W + (size_t)(nt * 16 + lm) * D_ATT + 2 * h;

    v8f c = {0.f, 0.f, 0.f, 0.f, 0.f, 0.f, 0.f, 0.f};

#pragma unroll 8
    for (int k0 = 0; k0 < D_ATT; k0 += 4) {
        v2f a = *(const v2f*)(arow + k0);        // {K=k0+2h, K=k0+2h+1}
        v2f b = *(const v2f*)(brow + k0);
        c = __builtin_amdgcn_wmma_f32_16x16x4_f32(
                /*neg_a=*/false, a, /*neg_b=*/false, b,
                /*c_mod=*/(short)0, c, /*reuse_a=*/false, /*reuse_b=*/false);
    }

    // C layout: VGPR r -> M = r + 8*h, N = lane&15
    float* __restrict__ o = Y + (size_t)(mt * 16 + 8 * h) * D_ATT + nt * 16 + lm;
#pragma unroll
    for (int r = 0; r < 8; ++r)
        o[(size_t)r * D_ATT] = c[r];
}

// ---------------------------------------------------------------------------
// Kernel 2: fused additive attention per (b, g, 16 query rows).
//   phase 1: VALU scores relu(sum_d relu(Qd-Kd)*w + bias); K chunks are
//            DOUBLE-BUFFERED through LDS with async global->LDS b128 copies
//            (ASYNCcnt path) so the next chunk streams in under the VALU work.
//   phase 2: row softmax (one thread per query row).
//   phase 3: P @ Vg with f32 WMMA; wave w owns D-columns [16w,16w+16).
// 64 threads (2 waves); grid = B * G * (L/16) = 1024 blocks.
// ---------------------------------------------------------------------------
#define QROWS   16
#define KCHUNK  64
#define NCHUNK  (SEQ / KCHUNK)    // 8

__global__ __launch_bounds__(64) void attn_fused_wmma(
    const float* __restrict__ Q,
    const float* __restrict__ K,
    const float* __restrict__ V,
    const float* __restrict__ w_mlp,
    const float* __restrict__ b_mlp,
    float* __restrict__ out)
{
    __shared__ float sScore[QROWS][SEQ];        // 32 KB
    __shared__ float sQ[QROWS][D_GROUP];        //  2 KB
    __shared__ float sK[2][KCHUNK][D_GROUP];    // 16 KB, double-buffered
    __shared__ float sW[D_GROUP];

    const int tid  = threadIdx.x;               // 0..63
    const int lane = tid & 31;
    const int wave = tid >> 5;

    const int blk = blockIdx.x;
    const int qt  = blk & 31;                   // 32 q-tiles
    const int g   = (blk >> 5) & 7;             // 8 groups
    const int b   = blk >> 8;                   // 4 batches
    const int q0  = qt * QROWS;

    const float bias = b_mlp[0];
    if (tid < D_GROUP) sW[tid] = w_mlp[tid];

    // stage Q tile (16 x 32)
    for (int i = tid; i < QROWS * D_GROUP; i += 64) {
        const int r = i >> 5, d = i & 31;
        sQ[r][d] = Q[(size_t)(b * SEQ + q0 + r) * D_ATT + g * D_GROUP + d];
    }

    // ---- phase 1: scores with async double-buffered K staging ----
    const float* __restrict__ kbase = K + (size_t)(b * SEQ) * D_ATT + g * D_GROUP;

    // prologue: async-stage chunk 0 into buffer 0
    for (int j = tid; j < KCHUNK * (D_GROUP / 4); j += 64) {
        const int r  = j >> 3;
        const int qd = (j & 7) * 4;
        const uint64_t ga = (uint64_t)(uintptr_t)(kbase + (size_t)r * D_ATT + qd);
        const uint32_t la = (uint32_t)(uintptr_t)&sK[0][r][qd];
        asm volatile("global_load_async_to_lds_b128 %0, %1, off"
                     :: "v"(la), "v"(ga) : "memory");
    }
    asm volatile("s_wait_asynccnt 0x0" ::: "memory");
    __syncthreads();

    for (int ci = 0; ci < NCHUNK; ++ci) {
        const int buf = ci & 1;

        // prefetch next chunk into the other buffer while we compute
        if (ci + 1 < NCHUNK) {
            const float* kb = kbase + (size_t)(ci + 1) * KCHUNK * D_ATT;
            for (int j = tid; j < KCHUNK * (D_GROUP / 4); j += 64) {
                const int r  = j >> 3;
                const int qd = (j & 7) * 4;
                const uint64_t ga = (uint64_t)(uintptr_t)(kb + (size_t)r * D_ATT + qd);
                const uint32_t la = (uint32_t)(uintptr_t)&sK[buf ^ 1][r][qd];
                asm volatile("global_load_async_to_lds_b128 %0, %1, off"
                             :: "v"(la), "v"(ga) : "memory");
            }
        }

        // score block: 16 q x 64 k against current buffer
        const int kc = ci * KCHUNK;
        for (int i = tid; i < QROWS * KCHUNK; i += 64) {
            const int q = i / KCHUNK, k = i % KCHUNK;
            float s = 0.f;
#pragma unroll
            for (int d = 0; d < D_GROUP; ++d)
                s = fmaf(fmaxf(sQ[q][d] - sK[buf][k][d], 0.f), sW[d], s);
            sScore[q][kc + k] = fmaxf(s + bias, 0.f);
        }

        // prefetch landed + everyone done with this buffer before reuse
        asm volatile("s_wait_asynccnt 0x0" ::: "memory");
        __syncthreads();
    }

    // ---- phase 2: softmax over k, one thread per query row ----
    if (tid < QROWS) {
        float mx = -3.402823e38f;
        for (int k = 0; k < SEQ; ++k) mx = fmaxf(mx, sScore[tid][k]);
        float sum = 0.f;
        for (int k = 0; k < SEQ; ++k) {
            const float e = __expf(sScore[tid][k] - mx);
            sScore[tid][k] = e;
            sum += e;
        }
        const float inv = 1.f / sum;
        for (int k = 0; k < SEQ; ++k) sScore[tid][k] *= inv;
    }
    __syncthreads();

    // ---- phase 3: out[b,q,d,g] = sum_k P[q,k] * V[b,k,d*8+g] via WMMA ----
    const int d0 = wave * 16;                   // this wave's D-tile
    const int lm = lane & 15;
    const int h  = lane >> 4;

    // B-frag: lane holds column n=lm -> V column (d0+lm)*8 + g, rows = k
    const float* __restrict__ vcol =
        V + (size_t)(b * SEQ) * D_ATT + (d0 + lm) * N_GROUP + g;

    v8f c = {0.f, 0.f, 0.f, 0.f, 0.f, 0.f, 0.f, 0.f};

#pragma unroll 4
    for (int k0 = 0; k0 < SEQ; k0 += 4) {
        const int ka = k0 + 2 * h;              // even -> 8B-aligned LDS read
        v2f a = *(const v2f*)&sScore[lm][ka];
        v2f bb;
        bb.x = vcol[(size_t)ka * D_ATT];
        bb.y = vcol[(size_t)(ka + 1) * D_ATT];
        c = __builtin_amdgcn_wmma_f32_16x16x4_f32(
                false, a, false, bb, (short)0, c, false, false);
    }

    // store: out flat index ((b*SEQ + q)*32 + d)*8 + g, q = q0 + r + 8h
    float* __restrict__ o =
        out + ((size_t)(b * SEQ + q0 + 8 * h) * D_GROUP + d0 + lm) * N_GROUP + g;
#pragma unroll
    for (int r = 0; r < 8; ++r)
        o[(size_t)r * D_GROUP * N_GROUP] = c[r];
}

// ---------------------------------------------------------------------------
extern "C" void kernel_launch(void* const* d_in, const int* in_sizes, int n_in,
                              void* d_out, int out_size, void* d_ws, size_t ws_size,
                              hipStream_t stream) {
    const float* x     = (const float*)d_in[0];
    const float* Wq    = (const float*)d_in[1];
    const float* Wk    = (const float*)d_in[2];
    const float* Wv    = (const float*)d_in[3];
    const float* w_mlp = (const float*)d_in[4];
    const float* b_mlp = (const float*)d_in[5];

    float* Qb = (float*)d_ws;                 // 3 x 2 MB scratch
    float* Kb = Qb + (size_t)M_TOT * D_ATT;
    float* Vb = Kb + (size_t)M_TOT * D_ATT;

    // 3 mats * 128 mtiles * 16 ntiles = 6144 waves / 4 waves per block
    qkv_proj_wmma<<<1536, 128, 0, stream>>>(x, Wq, Wk, Wv, Qb, Kb, Vb);

    // B * G * (SEQ/16) = 4*8*32 = 1024 blocks of 2 waves
    attn_fused_wmma<<<1024, 64, 0, stream>>>(Qb, Kb, Vb, w_mlp, b_mlp,
                                             (float*)d_out);
}